// EngramModule_15118284882110
// MI455X (gfx1250) — compile-verified
//
#include <hip/hip_runtime.h>
#include <hip/hip_bf16.h>
#include <math.h>

// ---------------- constants from the reference ----------------
#define B_      4
#define S_      2048
#define HID_    2048
#define EMB_    64
#define ORDERS_ 3
#define HEADS_  4
#define MEM_    768            // ORDERS*HEADS*EMB
#define TABLE_  200000
#define KCONV_  3
#define NTOK_   (B_ * S_)      // 8192

typedef float v2f __attribute__((ext_vector_type(2)));
typedef float v8f __attribute__((ext_vector_type(8)));

// ---------------- kernel 1: hash + gather ----------------
__global__ void hash_gather_kernel(const int* __restrict__ ids,
                                   const float* __restrict__ tables,
                                   float* __restrict__ mem) {
  const unsigned MULTS[4] = {2654435761u, 2246822519u, 3266489917u, 668265263u};
  const int t = blockIdx.x;            // token index 0..8191
  const int b = t >> 11;               // /S_
  const int s = t & (S_ - 1);
  const int* row = ids + (size_t)b * S_;
  float* mrow = mem + (size_t)t * MEM_;

  for (int i = threadIdx.x; i < MEM_; i += blockDim.x) {
    const int oh = i >> 6;             // (order,head) pair 0..11
    const int o  = oh >> 2;            // order index 0..2  (n = o+2)
    const int h  = oh & 3;
    const int n  = o + 2;
    unsigned acc = 0u;
    #pragma unroll
    for (int j = 0; j < 4; ++j) {      // at most MAX_N=4 terms
      if (j < n) {
        const int pos = s - (n - 1) + j;
        const unsigned v = (pos >= 0) ? (unsigned)row[pos] : 0u;
        acc = acc * 1000003u + v;
      }
    }
    const unsigned idx = (acc * MULTS[h]) % 200000u;
    mrow[i] = tables[((size_t)oh * TABLE_ + idx) * EMB_ + (i & 63)];
  }
}

// ---------------- kernel 2/4: f32 WMMA GEMM, 16x64 tile per wave ----------------
// C[M x N] = A[M x Kdim] * B[Kdim x N], all row-major f32.
// Block = 256 threads = 8 waves stacked along M -> block tile 128 x 64.
// Each wave: one A fragment (16x4) reused across 4 WMMAs covering N-span 64.
// A-frag (16x4 f32, 2 VGPR): lanes 0-15 hold K={k,k+1}, lanes 16-31 K={k+2,k+3}.
// B-frag (4x16 f32, 2 VGPR): same K split; N = n0 + j*16 + (lane&15).
// Optional per-output-row scale (alpha gating for the value GEMM).
__global__ void gemm_wmma_f32(const float* __restrict__ A,
                              const float* __restrict__ Bm,
                              float* __restrict__ C,
                              int N, int Kdim,
                              const float* __restrict__ rowscale) {
  const int wid  = threadIdx.x >> 5;     // 0..7 along M
  const int lane = threadIdx.x & 31;
  const bool hi  = lane >= 16;
  const int  l   = lane & 15;

  const int m0 = (blockIdx.y * 8 + wid) * 16;
  const int n0 = blockIdx.x * 64;

  const float* arow  = A  + (size_t)(m0 + l) * Kdim + (hi ? 2 : 0);
  // base for this lane's B column within the 64-wide tile (tile j adds j*16)
  const float* bbase = Bm + (size_t)(hi ? 2 * (size_t)N : 0) + n0 + l;

  v8f acc0 = {0.f,0.f,0.f,0.f,0.f,0.f,0.f,0.f};
  v8f acc1 = acc0, acc2 = acc0, acc3 = acc0;

  #pragma unroll 2
  for (int k = 0; k < Kdim; k += 4) {
    // A: two consecutive K values -> one 8-byte load, shared by 4 WMMAs
    v2f a = *(const v2f*)(arow + k);
    // WGP-scope prefetch one cacheline ahead on the A stream
    __builtin_prefetch(arow + k + 32, 0, 3);

    const float* bk0 = bbase + (size_t)k * N;      // K row (k or k+2)
    const float* bk1 = bk0 + N;                    // K row (k+1 or k+3)
    v2f b0; b0.x = bk0[ 0]; b0.y = bk1[ 0];
    v2f b1; b1.x = bk0[16]; b1.y = bk1[16];
    v2f b2; b2.x = bk0[32]; b2.y = bk1[32];
    v2f b3; b3.x = bk0[48]; b3.y = bk1[48];

    acc0 = __builtin_amdgcn_wmma_f32_16x16x4_f32(false, a, false, b0, (short)0, acc0, false, false);
    acc1 = __builtin_amdgcn_wmma_f32_16x16x4_f32(false, a, false, b1, (short)0, acc1, false, false);
    acc2 = __builtin_amdgcn_wmma_f32_16x16x4_f32(false, a, false, b2, (short)0, acc2, false, false);
    acc3 = __builtin_amdgcn_wmma_f32_16x16x4_f32(false, a, false, b3, (short)0, acc3, false, false);
  }

  const int rbase = m0 + (hi ? 8 : 0);
  #pragma unroll
  for (int r = 0; r < 8; ++r) {
    const int orow = rbase + r;
    const float sc = rowscale ? rowscale[orow] : 1.0f;
    float* crow = C + (size_t)orow * N + n0 + l;
    crow[ 0] = acc0[r] * sc;
    crow[16] = acc1[r] * sc;
    crow[32] = acc2[r] * sc;
    crow[48] = acc3[r] * sc;
  }
}

// ---------------- kernel 3: gate alpha ----------------
__global__ void alpha_kernel(const float* __restrict__ q,
                             const float* __restrict__ mem,
                             float* __restrict__ alpha) {
  const int t = blockIdx.x;
  const float* qr = q   + (size_t)t * MEM_;
  const float* mr = mem + (size_t)t * MEM_;
  float s = 0.f;
  for (int i = threadIdx.x; i < MEM_; i += blockDim.x) s += qr[i] * mr[i];
  __shared__ float red[256];
  red[threadIdx.x] = s;
  __syncthreads();
  for (int off = 128; off > 0; off >>= 1) {
    if (threadIdx.x < off) red[threadIdx.x] += red[threadIdx.x + off];
    __syncthreads();
  }
  if (threadIdx.x == 0) {
    const float x = red[0] * 0.036084391824351615f;   // 1/sqrt(768)
    alpha[t] = 1.f / (1.f + expf(-x));
  }
}

// ---------------- kernel 5: causal depthwise conv + bias + residual ----------------
__global__ void conv_residual_kernel(const float* __restrict__ hs,
                                     const float* __restrict__ gated,
                                     const float* __restrict__ w,
                                     const float* __restrict__ bias,
                                     float* __restrict__ out) {
  const size_t i = (size_t)blockIdx.x * blockDim.x + threadIdx.x;
  if (i >= (size_t)NTOK_ * HID_) return;
  const int h = (int)(i & (HID_ - 1));
  const size_t tok = i >> 11;              // /HID_
  const int s = (int)(tok & (S_ - 1));
  float acc = bias[h];
  #pragma unroll
  for (int k = 0; k < KCONV_; ++k) {
    const int sp = s - (KCONV_ - 1) + k;
    if (sp >= 0) acc += w[h * KCONV_ + k] * gated[i + ((size_t)(k - (KCONV_ - 1))) * HID_];
  }
  out[i] = hs[i] + acc;
}

// ---------------- launcher ----------------
extern "C" void kernel_launch(void* const* d_in, const int* in_sizes, int n_in,
                              void* d_out, int out_size, void* d_ws, size_t ws_size,
                              hipStream_t stream) {
  const float* hs     = (const float*)d_in[0];
  const int*   ids    = (const int*)  d_in[1];
  const float* tables = (const float*)d_in[2];
  const float* Wq     = (const float*)d_in[3];
  const float* Wv     = (const float*)d_in[4];
  const float* convw  = (const float*)d_in[5];
  const float* convb  = (const float*)d_in[6];
  float* out = (float*)d_out;

  float* mem   = (float*)d_ws;                         // NTOK_*MEM_
  float* q     = mem + (size_t)NTOK_ * MEM_;           // NTOK_*MEM_
  float* alpha = q   + (size_t)NTOK_ * MEM_;           // NTOK_
  float* gated = alpha + NTOK_;                        // NTOK_*HID_

  // 1) hash + gather memory rows
  hash_gather_kernel<<<NTOK_, 256, 0, stream>>>(ids, tables, mem);

  // 2) q = hs @ Wq    (M=8192, N=768, K=2048); block tile 128x64
  {
    dim3 grid(MEM_ / 64, NTOK_ / 128);                 // (12, 64)
    gemm_wmma_f32<<<grid, 256, 0, stream>>>(hs, Wq, q, MEM_, HID_, nullptr);
  }

  // 3) alpha = sigmoid(dot(q, mem)/sqrt(MEM))
  alpha_kernel<<<NTOK_, 256, 0, stream>>>(q, mem, alpha);

  // 4) gated = alpha * (mem @ Wv)   (M=8192, N=2048, K=768); block tile 128x64
  {
    dim3 grid(HID_ / 64, NTOK_ / 128);                 // (32, 64)
    gemm_wmma_f32<<<grid, 256, 0, stream>>>(mem, Wv, gated, HID_, MEM_, alpha);
  }

  // 5) causal depthwise conv (K=3) + bias + residual
  {
    const size_t total = (size_t)NTOK_ * HID_;
    conv_residual_kernel<<<(unsigned)((total + 255) / 256), 256, 0, stream>>>(
        hs, gated, convw, convb, out);
  }
}